// AutoRegressiveDecoderLayer_80376017977629
// MI455X (gfx1250) — compile-verified
//
#include <hip/hip_runtime.h>
#include <hip/hip_bf16.h>

// ---------------------------------------------------------------------------
// AutoRegressiveDecoderLayer for MI455X (gfx1250, wave32).
// Bandwidth-bound workload (~1.5 GB of K/V streamed once vs ~1 GFLOP), so:
//  * attention kernels = coalesced nontemporal streaming + LDS score rows
//  * dense (B,D)x(D,D) GEMMs = V_WMMA_F32_16X16X4_F32 (fp32-exact)
// ---------------------------------------------------------------------------

typedef float v2f __attribute__((ext_vector_type(2)));
typedef float v4f __attribute__((ext_vector_type(4)));
typedef float v8f __attribute__((ext_vector_type(8)));

#define B_  256
#define D_  256
#define L_  1024
#define N_  2048
#define H_  8
#define DH_ 32
#define BD_ (B_ * D_)

// ---------------------------------------------------------------------------
// GEMM: Y(M=256 x N=256) = X(256x256) @ W(256x256) + bias  [+ optional ReLU]
// One wave per 16x64 tile. grid = (M/16, N/64), block = 32 (full wave ->
// EXEC all ones, required by WMMA).
// Fragment layouts per CDNA5 ISA 7.12.2:
//   A 16x4 : lane m=l&15; VGPR0/1 hold K=(0,1) for lanes 0-15, K=(2,3) for 16-31
//   B 4x16 : lane n=l&15; VGPR0/1 hold K rows (0,1) / (2,3) analogously
//   C/D    : VGPR r -> (M=r, N=l) lanes 0-15, (M=r+8, N=l-16) lanes 16-31
// ---------------------------------------------------------------------------
__global__ __launch_bounds__(32) void gemm_bias_kernel(
    const float* __restrict__ X, const float* __restrict__ W,
    const float* __restrict__ bias, float* __restrict__ Y, int relu) {
  const int lane = threadIdx.x;            // 0..31
  const int row0 = blockIdx.x * 16;
  const int col0 = blockIdx.y * 64;
  const int m    = lane & 15;
  const int khi  = (lane >> 4) * 2;        // 0 for lanes 0-15, 2 for 16-31

  v8f acc0 = {}, acc1 = {}, acc2 = {}, acc3 = {};
  const float* xrow = X + (size_t)(row0 + m) * D_;

  for (int k = 0; k < D_; k += 4) {
    v2f a;
    a.x = xrow[k + khi];
    a.y = xrow[k + khi + 1];
    const float* w0 = W + (size_t)(k + khi) * D_;
    const float* w1 = W + (size_t)(k + khi + 1) * D_;
    const int nb = col0 + (lane & 15);
    v2f b;
    b.x = w0[nb +  0]; b.y = w1[nb +  0];
    acc0 = __builtin_amdgcn_wmma_f32_16x16x4_f32(false, a, false, b, (short)0, acc0, false, false);
    b.x = w0[nb + 16]; b.y = w1[nb + 16];
    acc1 = __builtin_amdgcn_wmma_f32_16x16x4_f32(false, a, false, b, (short)0, acc1, false, false);
    b.x = w0[nb + 32]; b.y = w1[nb + 32];
    acc2 = __builtin_amdgcn_wmma_f32_16x16x4_f32(false, a, false, b, (short)0, acc2, false, false);
    b.x = w0[nb + 48]; b.y = w1[nb + 48];
    acc3 = __builtin_amdgcn_wmma_f32_16x16x4_f32(false, a, false, b, (short)0, acc3, false, false);
  }

  const int rbase = row0 + ((lane < 16) ? 0 : 8);
  v8f accs[4] = {acc0, acc1, acc2, acc3};
#pragma unroll
  for (int nt = 0; nt < 4; ++nt) {
    const int n  = col0 + nt * 16 + (lane & 15);
    const float bv = bias[n];
#pragma unroll
    for (int r = 0; r < 8; ++r) {
      float v = accs[nt][r] + bv;
      if (relu) v = fmaxf(v, 0.0f);
      Y[(size_t)(rbase + r) * D_ + n] = v;
    }
  }
}

__device__ __forceinline__ float wave_max(float v) {
#pragma unroll
  for (int off = 16; off; off >>= 1) v = fmaxf(v, __shfl_xor(v, off, 32));
  return v;
}
__device__ __forceinline__ float wave_sum(float v) {
#pragma unroll
  for (int off = 16; off; off >>= 1) v += __shfl_xor(v, off, 32);
  return v;
}

// ---------------------------------------------------------------------------
// Self-attention: 1 query vs (L_cache + 1) keys, 8 heads.
// grid = B, block = 256 (8 waves). Scores live in LDS (8 x 1025 fp32).
// Phase 1: thread (h=t&7, j=t>>3 strided) -> dot(q[h], K[j]) via 128B/lane
//          nontemporal chunks (each wave-iteration covers a contiguous 4KB).
// Phase 2: wave w == head w: softmax reduce, then weighted-V with lane==dh
//          element -> perfectly coalesced 128B per wave per key.
// ---------------------------------------------------------------------------
__global__ __launch_bounds__(256) void self_attn_kernel(
    const float* __restrict__ q_s, const float* __restrict__ Kc,
    const float* __restrict__ Vc, const float* __restrict__ k_new,
    const float* __restrict__ v_new, float* __restrict__ out) {
  const int b   = blockIdx.x;
  const int tid = threadIdx.x;
  __shared__ float qs[D_];
  __shared__ float sc[H_][L_ + 32];   // 1025 used, padded

  qs[tid] = q_s[(size_t)b * D_ + tid];
  __syncthreads();

  const int h      = tid & 7;
  const int jstart = tid >> 3;        // 0..31
  const float scale = 0.17677669529663687f; // 1/sqrt(32)

  float qreg[DH_];
#pragma unroll
  for (int d = 0; d < DH_; ++d) qreg[d] = qs[h * DH_ + d];

  const float* Kbase = Kc + (size_t)b * L_ * D_;
  for (int j = jstart; j < L_; j += 32) {
    const v4f* kr = (const v4f*)(Kbase + (size_t)j * D_ + h * DH_);
    float acc = 0.0f;
#pragma unroll
    for (int c = 0; c < 8; ++c) {
      v4f kv = __builtin_nontemporal_load(kr + c);
      acc += qreg[c * 4 + 0] * kv.x + qreg[c * 4 + 1] * kv.y +
             qreg[c * 4 + 2] * kv.z + qreg[c * 4 + 3] * kv.w;
    }
    sc[h][j] = acc * scale;
  }
  if (jstart == 0) {                  // appended key (this step's k)
    const float* kn = k_new + (size_t)b * D_ + h * DH_;
    float acc = 0.0f;
#pragma unroll
    for (int d = 0; d < DH_; ++d) acc += qreg[d] * kn[d];
    sc[h][L_] = acc * scale;
  }
  __syncthreads();

  const int wave = tid >> 5;          // == head
  const int lane = tid & 31;
  float m = -1e30f;
  for (int j = lane; j < L_ + 1; j += 32) m = fmaxf(m, sc[wave][j]);
  m = wave_max(m);
  float s = 0.0f;
  for (int j = lane; j < L_ + 1; j += 32) {
    float e = __expf(sc[wave][j] - m);
    sc[wave][j] = e;
    s += e;
  }
  const float inv = 1.0f / wave_sum(s);

  const float* Vb = Vc + (size_t)b * L_ * D_ + wave * DH_ + lane;
  float a0 = 0, a1 = 0, a2 = 0, a3 = 0;
  for (int j = 0; j < L_; j += 4) {
    a0 += sc[wave][j + 0] * __builtin_nontemporal_load(Vb + (size_t)(j + 0) * D_);
    a1 += sc[wave][j + 1] * __builtin_nontemporal_load(Vb + (size_t)(j + 1) * D_);
    a2 += sc[wave][j + 2] * __builtin_nontemporal_load(Vb + (size_t)(j + 2) * D_);
    a3 += sc[wave][j + 3] * __builtin_nontemporal_load(Vb + (size_t)(j + 3) * D_);
  }
  float acc = a0 + a1 + a2 + a3;
  acc += sc[wave][L_] * v_new[(size_t)b * D_ + wave * DH_ + lane];
  out[(size_t)b * D_ + wave * DH_ + lane] = acc * inv;
}

// ---------------------------------------------------------------------------
// Masked cross-attention: 1 query vs N=2048 keys. grid = (B, 2): each block
// handles 4 heads (keeps LDS scores at 4x2048 fp32 = 33KB). block = 128.
// Masked keys skip the K read and have weight exactly 0 -> V read skipped.
// ---------------------------------------------------------------------------
__global__ __launch_bounds__(128) void cross_attn_kernel(
    const float* __restrict__ q, const float* __restrict__ K,
    const float* __restrict__ V, const unsigned char* __restrict__ mask,
    float* __restrict__ out) {
  const int b   = blockIdx.x;
  const int hg  = blockIdx.y;         // head group: heads hg*4 .. hg*4+3
  const int tid = threadIdx.x;        // 0..127
  __shared__ float qs[128];
  __shared__ float sc[4][N_ + 8];

  qs[tid] = q[(size_t)b * D_ + hg * 128 + tid];
  __syncthreads();

  const int hl     = tid & 3;
  const int jstart = tid >> 2;        // 0..31
  const float scale = 0.17677669529663687f;

  float qreg[DH_];
#pragma unroll
  for (int d = 0; d < DH_; ++d) qreg[d] = qs[hl * DH_ + d];

  const float* Kbase = K + (size_t)b * N_ * D_ + (hg * 4 + hl) * DH_;
  const unsigned char* mrow = mask + (size_t)b * N_;
  for (int j = jstart; j < N_; j += 32) {
    float sv;
    if (mrow[j]) {
      sv = -1e9f;
    } else {
      const v4f* kr = (const v4f*)(Kbase + (size_t)j * D_);
      float acc = 0.0f;
#pragma unroll
      for (int c = 0; c < 8; ++c) {
        v4f kv = __builtin_nontemporal_load(kr + c);
        acc += qreg[c * 4 + 0] * kv.x + qreg[c * 4 + 1] * kv.y +
               qreg[c * 4 + 2] * kv.z + qreg[c * 4 + 3] * kv.w;
      }
      sv = acc * scale;
    }
    sc[hl][j] = sv;
  }
  __syncthreads();

  const int wave = tid >> 5;          // local head 0..3
  const int lane = tid & 31;
  float m = -1e30f;
  for (int j = lane; j < N_; j += 32) m = fmaxf(m, sc[wave][j]);
  m = wave_max(m);
  float s = 0.0f;
  for (int j = lane; j < N_; j += 32) {
    float e = __expf(sc[wave][j] - m);
    sc[wave][j] = e;
    s += e;
  }
  const float inv = 1.0f / wave_sum(s);

  const float* Vb = V + (size_t)b * N_ * D_ + (hg * 4 + wave) * DH_ + lane;
  float a0 = 0, a1 = 0, a2 = 0, a3 = 0;
  for (int j = 0; j < N_; j += 4) {
    float w0 = sc[wave][j + 0], w1 = sc[wave][j + 1];
    float w2 = sc[wave][j + 2], w3 = sc[wave][j + 3];
    if (w0 != 0.0f) a0 += w0 * __builtin_nontemporal_load(Vb + (size_t)(j + 0) * D_);
    if (w1 != 0.0f) a1 += w1 * __builtin_nontemporal_load(Vb + (size_t)(j + 1) * D_);
    if (w2 != 0.0f) a2 += w2 * __builtin_nontemporal_load(Vb + (size_t)(j + 2) * D_);
    if (w3 != 0.0f) a3 += w3 * __builtin_nontemporal_load(Vb + (size_t)(j + 3) * D_);
  }
  out[(size_t)b * D_ + (hg * 4 + wave) * DH_ + lane] = (a0 + a1 + a2 + a3) * inv;
}

// ---------------------------------------------------------------------------
// out = LayerNorm(x + y) * g + beta  over D=256. grid = B, block = 256.
// ---------------------------------------------------------------------------
__global__ __launch_bounds__(256) void resid_ln_kernel(
    const float* __restrict__ x, const float* __restrict__ y,
    const float* __restrict__ g, const float* __restrict__ beta,
    float* __restrict__ out) {
  const int b = blockIdx.x, t = threadIdx.x;
  __shared__ float red[8];
  __shared__ float stat;

  const float v = x[(size_t)b * D_ + t] + y[(size_t)b * D_ + t];

  float s = wave_sum(v);
  if ((t & 31) == 0) red[t >> 5] = s;
  __syncthreads();
  if (t == 0) {
    float m = 0;
#pragma unroll
    for (int i = 0; i < 8; ++i) m += red[i];
    stat = m * (1.0f / D_);
  }
  __syncthreads();
  const float mu = stat;
  const float dv = v - mu;
  s = wave_sum(dv * dv);
  __syncthreads();
  if ((t & 31) == 0) red[t >> 5] = s;
  __syncthreads();
  if (t == 0) {
    float m = 0;
#pragma unroll
    for (int i = 0; i < 8; ++i) m += red[i];
    stat = m * (1.0f / D_);
  }
  __syncthreads();
  const float var = stat;
  out[(size_t)b * D_ + t] = dv * rsqrtf(var + 1e-5f) * g[t] + beta[t];
}

// ---------------------------------------------------------------------------
extern "C" void kernel_launch(void* const* d_in, const int* in_sizes, int n_in,
                              void* d_out, int out_size, void* d_ws, size_t ws_size,
                              hipStream_t stream) {
  const float* h_t     = (const float*)d_in[0];
  const float* K_att   = (const float*)d_in[1];
  const float* V_att   = (const float*)d_in[2];
  const float* K_cache = (const float*)d_in[3];
  const float* V_cache = (const float*)d_in[4];
  const unsigned char* mask = (const unsigned char*)d_in[5];
  const float* wq_s = (const float*)d_in[6];   const float* bq_s = (const float*)d_in[7];
  const float* wk_s = (const float*)d_in[8];   const float* bk_s = (const float*)d_in[9];
  const float* wv_s = (const float*)d_in[10];  const float* bv_s = (const float*)d_in[11];
  const float* w0_s = (const float*)d_in[12];  const float* b0_s = (const float*)d_in[13];
  const float* wq_a = (const float*)d_in[14];  const float* bq_a = (const float*)d_in[15];
  const float* w0_a = (const float*)d_in[16];  const float* b0_a = (const float*)d_in[17];
  const float* w1   = (const float*)d_in[18];  const float* b1   = (const float*)d_in[19];
  const float* w2   = (const float*)d_in[20];  const float* b2   = (const float*)d_in[21];
  const float* ln1_g = (const float*)d_in[22]; const float* ln1_b = (const float*)d_in[23];
  const float* ln2_g = (const float*)d_in[24]; const float* ln2_b = (const float*)d_in[25];
  const float* ln3_g = (const float*)d_in[26]; const float* ln3_b = (const float*)d_in[27];

  float* ws   = (float*)d_ws;
  float* q_s  = ws + 0 * BD_;
  float* k_s  = ws + 1 * BD_;
  float* v_s  = ws + 2 * BD_;
  float* att1 = ws + 3 * BD_;
  float* t0   = ws + 4 * BD_;
  float* h1   = ws + 5 * BD_;
  float* q_a  = ws + 6 * BD_;
  float* att2 = ws + 7 * BD_;
  float* t1   = ws + 8 * BD_;
  float* h2   = ws + 9 * BD_;
  float* m1   = ws + 10 * BD_;
  float* m2   = ws + 11 * BD_;

  const dim3 ggrid(B_ / 16, D_ / 64);   // 16 x 4 tiles
  const dim3 gblk(32);

  // q/k/v projections
  gemm_bias_kernel<<<ggrid, gblk, 0, stream>>>(h_t, wq_s, bq_s, q_s, 0);
  gemm_bias_kernel<<<ggrid, gblk, 0, stream>>>(h_t, wk_s, bk_s, k_s, 0);
  gemm_bias_kernel<<<ggrid, gblk, 0, stream>>>(h_t, wv_s, bv_s, v_s, 0);

  // self-attention over cache + appended k/v
  self_attn_kernel<<<dim3(B_), dim3(256), 0, stream>>>(q_s, K_cache, V_cache, k_s, v_s, att1);
  gemm_bias_kernel<<<ggrid, gblk, 0, stream>>>(att1, w0_s, b0_s, t0, 0);
  resid_ln_kernel<<<dim3(B_), dim3(256), 0, stream>>>(h_t, t0, ln1_g, ln1_b, h1);

  // masked cross-attention
  gemm_bias_kernel<<<ggrid, gblk, 0, stream>>>(h1, wq_a, bq_a, q_a, 0);
  cross_attn_kernel<<<dim3(B_, 2), dim3(128), 0, stream>>>(q_a, K_att, V_att, mask, att2);
  gemm_bias_kernel<<<ggrid, gblk, 0, stream>>>(att2, w0_a, b0_a, t1, 0);
  resid_ln_kernel<<<dim3(B_), dim3(256), 0, stream>>>(h1, t1, ln2_g, ln2_b, h2);

  // MLP
  gemm_bias_kernel<<<ggrid, gblk, 0, stream>>>(h2, w1, b1, m1, 1);
  gemm_bias_kernel<<<ggrid, gblk, 0, stream>>>(m1, w2, b2, m2, 0);
  resid_ln_kernel<<<dim3(B_), dim3(256), 0, stream>>>(h2, m2, ln3_g, ln3_b, (float*)d_out);
}